// TorchMultiTokenAttention_50603304681981
// MI455X (gfx1250) — compile-verified
//
#include <hip/hip_runtime.h>
#include <hip/hip_bf16.h>

#define LL   2048
#define CC   32
#define PAD  2
#define TY   4
#define TX   48
#define HX   (TX + 4)   // 52 haloed cols
#define HY   (TY + 4)   // 8 haloed rows
#define CP   34         // padded channel stride in LDS (even -> 8B-aligned pairs)

typedef float v2f __attribute__((ext_vector_type(2)));
typedef float v8f __attribute__((ext_vector_type(8)));

// ---------------------------------------------------------------------------
// Pass 0: repack weight[32][32][5][5] into WMMA A-operand layout.
// Flat index f = (((mh*25 + tap)*8 + kk)*32 + lane)*2 + j  (25600 floats)
// lane 0-15 : M = lane,    holds K = 4*kk + {0,1}
// lane 16-31: M = lane-16, holds K = 4*kk + {2,3}
// ---------------------------------------------------------------------------
__global__ void wt_transform_kernel(const float* __restrict__ w,
                                    float* __restrict__ wt) {
    int f = blockIdx.x * blockDim.x + threadIdx.x;
    if (f >= 2 * 25 * 8 * 32 * 2) return;
    int j    = f & 1;
    int lane = (f >> 1) & 31;
    int kk   = (f >> 6) & 7;
    int tap  = (f >> 9) % 25;
    int mh   = f / (25 << 9);
    int co   = mh * 16 + (lane & 15);
    int k    = kk * 4 + ((lane >> 4) << 1) + j;        // input channel
    wt[f] = w[co * (CC * 25) + k * 25 + tap];
}

// ---------------------------------------------------------------------------
// Pass 1: per causal row (c, r): rowmax and 1/sum(exp(x - rowmax)) over [0,r].
// One 256-thread block per row.
// ---------------------------------------------------------------------------
__global__ __launch_bounds__(256)
void softmax_stats_kernel(const float* __restrict__ scores,
                          float* __restrict__ rmax,
                          float* __restrict__ rinv) {
    __shared__ float red[256];
    int row = blockIdx.x;              // c * LL + r
    int r   = row & (LL - 1);
    const float* s = scores + (size_t)row * LL;
    int tid = threadIdx.x;

    float m = -3.0e38f;
    for (int x = tid; x <= r; x += 256) m = fmaxf(m, s[x]);
    red[tid] = m;
    __syncthreads();
    for (int o = 128; o > 0; o >>= 1) {
        if (tid < o) red[tid] = fmaxf(red[tid], red[tid + o]);
        __syncthreads();
    }
    float rm = red[0];
    __syncthreads();

    float acc = 0.0f;
    for (int x = tid; x <= r; x += 256) acc += __expf(s[x] - rm);
    red[tid] = acc;
    __syncthreads();
    for (int o = 128; o > 0; o >>= 1) {
        if (tid < o) red[tid] += red[tid + o];
        __syncthreads();
    }
    if (tid == 0) { rmax[row] = rm; rinv[row] = 1.0f / red[0]; }
}

// ---------------------------------------------------------------------------
// Pass 2: recompute probs tile into LDS, then 5x5 x channel-mix conv as
// sum of 25 tap-GEMMs on v_wmma_f32_16x16x4_f32, bias + causal re-mask.
// Block: 256 threads (8 wave32). Tile: 32 co x TY rows x TX cols.
// Wave w owns M-half (w&1) and 3 N-groups -> A operands hoisted per tap
// and reused across the 3 quadrants (1 vmem b64 per 3 WMMAs).
// ---------------------------------------------------------------------------
__global__ __launch_bounds__(256)
void conv_wmma_kernel(const float* __restrict__ scores,
                      const float* __restrict__ wt,
                      const float* __restrict__ bias,
                      const float* __restrict__ rmax,
                      const float* __restrict__ rinv,
                      float* __restrict__ out) {
    __shared__ float P[HY * HX * CP];   // 8*52*34*4 = 56,576 B

    int x0  = blockIdx.x * TX;
    int y0  = blockIdx.y * TY;
    int tid = threadIdx.x;

    // ---- Phase A: stage haloed probability tile (channel-fastest layout) ----
    for (int i = tid; i < CC * HY * HX; i += 256) {      // 13312 = 52*256, uniform
        int xx = i % HX;
        int yy = (i / HX) % HY;
        int ci = i / (HX * HY);
        int y  = y0 - PAD + yy;
        int x  = x0 - PAD + xx;
        float p = 0.0f;
        if (y >= 0 && y < LL && x >= 0 && x <= y) {      // causal + bounds
            int row  = ci * LL + y;
            float sc = scores[(size_t)row * LL + x];
            p = __expf(sc - rmax[row]) * rinv[row];
        }
        P[(yy * HX + xx) * CP + ci] = p;
    }
    __syncthreads();

    // ---- Phase B: WMMA tap-GEMMs ----
    int lane  = tid & 31;
    int wave  = tid >> 5;
    int nlo   = lane & 15;       // N within quadrant
    int hi    = lane >> 4;       // lane half (K pair select / M+8 select)
    int mh    = wave & 1;        // this wave's M half (co 0-15 / 16-31)
    int gbase = (wave >> 1) * 3; // this wave's 3 N-groups out of 12
    const v2f* wt2 = (const v2f*)wt + (mh * 25 * 8) * 32 + lane;

    // per-quadrant geometry (uniform per wave)
    int tyq[3], nxq[3];
#pragma unroll
    for (int q = 0; q < 3; ++q) {
        int g  = gbase + q;
        tyq[q] = g / 3;
        nxq[q] = (g % 3) * 16;
    }

    // accumulators seeded with bias (C/D layout: VGPR j -> M=j (lo) / j+8 (hi))
    int cobase = mh * 16 + hi * 8;
    v8f acc[3];
#pragma unroll
    for (int q = 0; q < 3; ++q)
#pragma unroll
        for (int j = 0; j < 8; ++j) acc[q][j] = bias[cobase + j];

    for (int ky = 0; ky < 5; ++ky) {
        for (int kx = 0; kx < 5; ++kx) {
            int tap = ky * 5 + kx;
            // hoisted A operands for this tap (shared by all 3 quadrants)
            v2f a[8];
#pragma unroll
            for (int kk = 0; kk < 8; ++kk) a[kk] = wt2[(tap * 8 + kk) * 32];
#pragma unroll
            for (int q = 0; q < 3; ++q) {
                int yy = tyq[q] + ky;                    // haloed input row (0..7)
                int xx = nxq[q] + nlo + kx;              // haloed input col (0..51)
                const float* pb = &P[(yy * HX + xx) * CP + hi * 2];
#pragma unroll
                for (int kk = 0; kk < 8; ++kk) {
                    v2f b = *(const v2f*)(pb + kk * 4);  // ds_load_b64
                    acc[q] = __builtin_amdgcn_wmma_f32_16x16x4_f32(
                                 false, a[kk], false, b, (short)0, acc[q],
                                 false, false);
                }
            }
        }
    }

    // ---- store with causal re-mask ----
#pragma unroll
    for (int q = 0; q < 3; ++q) {
        int y = y0 + tyq[q];
        int x = x0 + nxq[q] + nlo;
        if (y < LL && x < LL) {
#pragma unroll
            for (int j = 0; j < 8; ++j) {
                int co = cobase + j;
                out[((size_t)co * LL + y) * LL + x] = (x <= y) ? acc[q][j] : 0.0f;
            }
        }
    }
}

// ---------------------------------------------------------------------------
extern "C" void kernel_launch(void* const* d_in, const int* in_sizes, int n_in,
                              void* d_out, int out_size, void* d_ws, size_t ws_size,
                              hipStream_t stream) {
    (void)in_sizes; (void)n_in; (void)out_size; (void)ws_size;
    const float* scores = (const float*)d_in[0];   // [1,32,2048,2048]
    const float* weight = (const float*)d_in[1];   // [32,32,5,5]
    const float* bias   = (const float*)d_in[2];   // [32]
    float* out = (float*)d_out;

    // workspace layout (floats): rowmax[C*L] | rowinv[C*L] | wt[25600]
    float* ws   = (float*)d_ws;
    float* rmax = ws;
    float* rinv = ws + CC * LL;
    float* wt   = ws + 2 * CC * LL;

    wt_transform_kernel<<<(25600 + 255) / 256, 256, 0, stream>>>(weight, wt);
    softmax_stats_kernel<<<CC * LL, 256, 0, stream>>>(scores, rmax, rinv);

    dim3 grid((LL + TX - 1) / TX, LL / TY);
    conv_wmma_kernel<<<grid, 256, 0, stream>>>(scores, wt, bias, rmax, rinv, out);
}